// MinLoss_15229954032079
// MI455X (gfx1250) — compile-verified
//
#include <hip/hip_runtime.h>
#include <math.h>

typedef __attribute__((ext_vector_type(2))) float v2f;
typedef __attribute__((ext_vector_type(8))) float v8f;

#define BS      8
#define SEQ     4096
#define NSRC    4
#define DIM     257
#define ROW     (NSRC * DIM)          // 1028 floats per time step
#define PER_B   (SEQ * ROW)           // 4210688 floats per batch per tensor
#define BPB     64                    // blocks per batch
#define WPB     8                     // waves per block (256 threads, wave32)
#define WAVES   (BPB * WPB)           // 512 waves per batch
#define CHUNKS  (SEQ * 16)            // 65536 chunks of 16 d-positions per batch
#define WSF     24                    // floats of ws per batch: pg[16], pp[4], gg[4]

__global__ void MinLoss_zero_ws(float* ws) {
    int i = threadIdx.x;
    if (i < BS * WSF) ws[i] = 0.0f;
}

// Main streaming Gram kernel: one WMMA per 16 d-positions x 4 sources.
__global__ __launch_bounds__(256) void MinLoss_gram_main(
    const float* __restrict__ P, const float* __restrict__ G, float* __restrict__ ws)
{
    __shared__ float lds[WSF];
    const int tid = threadIdx.x;
    if (tid < WSF) lds[tid] = 0.0f;
    __syncthreads();

    const int b    = blockIdx.y;
    const int wave = blockIdx.x * WPB + (tid >> 5);   // 0..511 within batch
    const int lane = tid & 31;
    const int hi   = lane >> 4;        // 0: K=0,1   1: K=2,3
    const int lp   = lane & 15;        // M (for A) == N (for B)
    const int s    = lp & 3;           // source index
    const int u    = lp >> 2;          // k-slice 0..3

    // First chunk for this wave; chunk c -> (t = c>>4, dbase = (c&15)*16).
    const int t0    = wave >> 4;
    const int dbase = (wave & 15) << 4;
    const int doff  = 4 * u + 2 * hi;  // lane's first d within the 16-chunk
    long off = (long)b * PER_B + (long)t0 * ROW + s * DIM + dbase + doff;
    const float* pP = P + off;
    const float* pG = G + off;
    const long stride = (long)(WAVES >> 4) * ROW;     // 32 * 1028 floats per iter

    v8f acc = {};                       // 16x16 accumulator; diagonal blocks valid
    float accP = 0.0f, accG = 0.0f;     // per-lane norm partials for source s
    const int iters = CHUNKS / WAVES;   // 128, identical for every wave

    for (int i = 0; i < iters; ++i) {
        v2f a, bv;
        a.x  = pP[0]; a.y  = pP[1];
        bv.x = pG[0]; bv.y = pG[1];
        __builtin_prefetch(pP + 2 * stride, 0, 0);
        __builtin_prefetch(pG + 2 * stride, 0, 0);
        // D = A(16x4) * B(4x16) + C : v_wmma_f32_16x16x4_f32
        acc = __builtin_amdgcn_wmma_f32_16x16x4_f32(
            false, a, false, bv, (short)0, acc, false, false);
        accP = fmaf(a.x,  a.x,  accP); accP = fmaf(a.y,  a.y,  accP);
        accG = fmaf(bv.x, bv.x, accG); accG = fmaf(bv.y, bv.y, accG);
        pP += stride; pG += stride;
    }

    // Norms: lane covers source s.
    atomicAdd(&lds[16 + s], accP);
    atomicAdd(&lds[20 + s], accG);

    // Gram: lane holds column N=lp (t = lp&3, v = lp>>2). Diagonal u==v rows
    // s+4v live in this lane iff (lp>>3) == hi; they sit in VGPRs 4*((v)&1)+s.
    if ((lp >> 3) == hi) {
        const int tcol = lp & 3;
        const int vv   = (lp >> 2) & 1;
        #pragma unroll
        for (int ss = 0; ss < 4; ++ss)
            atomicAdd(&lds[ss * 4 + tcol], acc[4 * vv + ss]);
    }
    __syncthreads();
    if (tid < WSF) atomicAdd(&ws[b * WSF + tid], lds[tid]);
}

// Tail: the single leftover column d = 256 of each time step.
__global__ __launch_bounds__(256) void MinLoss_gram_tail(
    const float* __restrict__ P, const float* __restrict__ G, float* __restrict__ ws)
{
    __shared__ float lds[WSF];
    const int tid = threadIdx.x;
    if (tid < WSF) lds[tid] = 0.0f;
    __syncthreads();

    const int b = blockIdx.x;
    float pg[16] = {}, pp[4] = {}, gg[4] = {};
    for (int t = tid; t < SEQ; t += 256) {
        const long base = (long)b * PER_B + (long)t * ROW + (DIM - 1);
        float pv[4], gv[4];
        #pragma unroll
        for (int k = 0; k < 4; ++k) { pv[k] = P[base + k * DIM]; gv[k] = G[base + k * DIM]; }
        #pragma unroll
        for (int ss = 0; ss < 4; ++ss) {
            pp[ss] = fmaf(pv[ss], pv[ss], pp[ss]);
            gg[ss] = fmaf(gv[ss], gv[ss], gg[ss]);
            #pragma unroll
            for (int tt = 0; tt < 4; ++tt)
                pg[ss * 4 + tt] = fmaf(pv[ss], gv[tt], pg[ss * 4 + tt]);
        }
    }
    #pragma unroll
    for (int k = 0; k < 16; ++k) atomicAdd(&lds[k], pg[k]);
    #pragma unroll
    for (int k = 0; k < 4; ++k) { atomicAdd(&lds[16 + k], pp[k]); atomicAdd(&lds[20 + k], gg[k]); }
    __syncthreads();
    if (tid < WSF) atomicAdd(&ws[b * WSF + tid], lds[tid]);
}

// Tiny finalize: build d2, greedy match (S=4), sum matched distances / bs.
__global__ void MinLoss_finalize(const float* __restrict__ ws, float* __restrict__ out)
{
    if (threadIdx.x != 0 || blockIdx.x != 0) return;
    float total = 0.0f;
    for (int b = 0; b < BS; ++b) {
        const float* w = ws + b * WSF;
        float d2[16], dm[16];
        for (int ss = 0; ss < 4; ++ss)
            for (int tt = 0; tt < 4; ++tt) {
                float v = w[16 + ss] + w[20 + tt] - 2.0f * w[ss * 4 + tt];
                v = v > 0.0f ? v : 0.0f;   // clamp, as in reference
                d2[ss * 4 + tt] = v;
                dm[ss * 4 + tt] = v;       // sqrt is monotone: match on d2
            }
        int match[4] = {0, 0, 0, 0};
        for (int r = 0; r < 4; ++r) {
            int mi = 0; float mv = dm[0];
            for (int k = 1; k < 16; ++k) if (dm[k] < mv) { mv = dm[k]; mi = k; }
            const int rr = mi >> 2, cc = mi & 3;
            match[rr] = cc;
            for (int k = 0; k < 4; ++k) { dm[rr * 4 + k] = INFINITY; dm[k * 4 + cc] = INFINITY; }
        }
        for (int ss = 0; ss < 4; ++ss) total += sqrtf(d2[ss * 4 + match[ss]]);
    }
    out[0] = total / (float)BS;
}

extern "C" void kernel_launch(void* const* d_in, const int* in_sizes, int n_in,
                              void* d_out, int out_size, void* d_ws, size_t ws_size,
                              hipStream_t stream) {
    const float* P = (const float*)d_in[0];
    const float* G = (const float*)d_in[1];
    float* out = (float*)d_out;
    float* ws  = (float*)d_ws;   // needs BS*24 floats = 768 bytes

    MinLoss_zero_ws<<<1, 256, 0, stream>>>(ws);
    MinLoss_gram_main<<<dim3(BPB, BS), 256, 0, stream>>>(P, G, ws);
    MinLoss_gram_tail<<<BS, 256, 0, stream>>>(P, G, ws);
    MinLoss_finalize<<<1, 32, 0, stream>>>(ws, out);
}